// RouterNetwork_63513976373277
// MI455X (gfx1250) — compile-verified
//
#include <hip/hip_runtime.h>

// ---------------- constants (match reference) ----------------
constexpr int   D_DIM = 1024;
constexpr int   E_DIM = 64;
constexpr float EPS_F = 1e-8f;
constexpr float LBW   = 0.01f;

typedef __attribute__((ext_vector_type(2))) float v2f;
typedef __attribute__((ext_vector_type(8))) float v8f;

// ---------------- kernel T: transpose router weight W[D,E] -> WT[E,D] ----------------
__global__ __launch_bounds__(256)
void rn_transpose_w_kernel(const float* __restrict__ Wm, float* __restrict__ WT) {
    const int n = blockIdx.x * 256 + threadIdx.x;   // n in [0, D*E)
    const int k = n >> 6;          // D index
    const int e = n & 63;          // E index
    WT[(size_t)e * D_DIM + k] = Wm[(size_t)k * E_DIM + e];
}

// ---------------- kernel 0: init scratch accumulators ----------------
__global__ void rn_init_kernel(float* usage_sum, float* ent_sum, float* conf_sum) {
    int t = threadIdx.x;
    if (t < E_DIM)      usage_sum[t] = 0.0f;
    if (t == E_DIM)     ent_sum[0]   = 0.0f;
    if (t == E_DIM + 1) conf_sum[0]  = 0.0f;
}

// ---------------- kernel 1: WMMA GEMM + softmax + top-2 ----------------
// 128 threads (4 waves); each wave computes a 32-row x 64-expert logits panel
// as 2 M-tiles x 4 N-tiles of V_WMMA_F32_16X16X4_F32, reusing B fragments
// across the two M-tiles. 128 rows per block.
__global__ __launch_bounds__(128)
void rn_gemm_softmax_kernel(const float* __restrict__ H,     // [nTok, D]
                            const float* __restrict__ WT,    // [E, D] (transposed)
                            const float* __restrict__ bias,  // [E]
                            float* __restrict__ out_probs,   // d_out + OFF_PROBS  [nTok, E]
                            float* __restrict__ out_idx_f,   // d_out + 0          [nTok, 2]
                            int*   __restrict__ idx_i32,     // ws                 [nTok, 2]
                            float* __restrict__ p2,          // ws premask renorm top2
                            float* __restrict__ usage_sum,   // ws [E]
                            float* __restrict__ ent_sum)     // ws [1]
{
    __shared__ float sh_logits[4][32][E_DIM + 1];
    __shared__ float sh_usage[E_DIM];
    __shared__ float sh_ent;

    const int t    = threadIdx.x;
    const int wave = t >> 5;
    const int lane = t & 31;
    if (t < E_DIM) sh_usage[t] = 0.0f;
    if (t == 0)    sh_ent = 0.0f;

    const int row0  = (blockIdx.x * 4 + wave) * 32;   // first of 32 rows for this wave
    const int m     = lane & 15;                      // A row / B col within 16-block
    const int khalf = (lane >> 4) << 1;               // lanes 0-15: K=0,1  lanes 16-31: K=2,3

    // accumulators: [mtile][ntile]
    v8f acc00 = {}, acc01 = {}, acc02 = {}, acc03 = {};
    v8f acc10 = {}, acc11 = {}, acc12 = {}, acc13 = {};

    const float* __restrict__ arow0 = H + (size_t)(row0 + m) * D_DIM;
    const float* __restrict__ arow1 = arow0 + (size_t)16 * D_DIM;
    const float* __restrict__ wt0 = WT + (size_t)(m)      * D_DIM;
    const float* __restrict__ wt1 = WT + (size_t)(16 + m) * D_DIM;
    const float* __restrict__ wt2 = WT + (size_t)(32 + m) * D_DIM;
    const float* __restrict__ wt3 = WT + (size_t)(48 + m) * D_DIM;

    for (int k0 = 0; k0 < D_DIM; k0 += 4) {
        const int kk = k0 + khalf;                    // even -> 8B aligned
        const v2f a0 = *(const v2f*)(arow0 + kk);
        const v2f a1 = *(const v2f*)(arow1 + kk);
        const v2f b0 = *(const v2f*)(wt0 + kk);
        const v2f b1 = *(const v2f*)(wt1 + kk);
        const v2f b2 = *(const v2f*)(wt2 + kk);
        const v2f b3 = *(const v2f*)(wt3 + kk);
        acc00 = __builtin_amdgcn_wmma_f32_16x16x4_f32(false, a0, false, b0, (short)0, acc00, false, false);
        acc01 = __builtin_amdgcn_wmma_f32_16x16x4_f32(false, a0, false, b1, (short)0, acc01, false, false);
        acc02 = __builtin_amdgcn_wmma_f32_16x16x4_f32(false, a0, false, b2, (short)0, acc02, false, false);
        acc03 = __builtin_amdgcn_wmma_f32_16x16x4_f32(false, a0, false, b3, (short)0, acc03, false, false);
        acc10 = __builtin_amdgcn_wmma_f32_16x16x4_f32(false, a1, false, b0, (short)0, acc10, false, false);
        acc11 = __builtin_amdgcn_wmma_f32_16x16x4_f32(false, a1, false, b1, (short)0, acc11, false, false);
        acc12 = __builtin_amdgcn_wmma_f32_16x16x4_f32(false, a1, false, b2, (short)0, acc12, false, false);
        acc13 = __builtin_amdgcn_wmma_f32_16x16x4_f32(false, a1, false, b3, (short)0, acc13, false, false);
    }

    // C layout: VGPR j, lanes 0-15 -> M=j, lanes 16-31 -> M=j+8; N = lane%16 in 16-col block
    const int rofs = (lane < 16) ? 0 : 8;
#pragma unroll
    for (int j = 0; j < 8; ++j) {
        sh_logits[wave][rofs + j][ 0 + m] = acc00[j] + bias[ 0 + m];
        sh_logits[wave][rofs + j][16 + m] = acc01[j] + bias[16 + m];
        sh_logits[wave][rofs + j][32 + m] = acc02[j] + bias[32 + m];
        sh_logits[wave][rofs + j][48 + m] = acc03[j] + bias[48 + m];
        sh_logits[wave][16 + rofs + j][ 0 + m] = acc10[j] + bias[ 0 + m];
        sh_logits[wave][16 + rofs + j][16 + m] = acc11[j] + bias[16 + m];
        sh_logits[wave][16 + rofs + j][32 + m] = acc12[j] + bias[32 + m];
        sh_logits[wave][16 + rofs + j][48 + m] = acc13[j] + bias[48 + m];
    }
    __syncthreads();

    // one thread per token-row: softmax over 64 experts + top-2 (all 128 threads active)
    {
        const int w2 = t >> 5, r = t & 31;
        const int gRow = blockIdx.x * 128 + t;
        const float* L = &sh_logits[w2][r][0];
        float mx = -3.4e38f;
        for (int e = 0; e < E_DIM; ++e) mx = fmaxf(mx, L[e]);
        float s = 0.0f;
        for (int e = 0; e < E_DIM; ++e) s += __expf(L[e] - mx);
        const float inv = 1.0f / s;

        float v0 = -1.0f, v1 = -1.0f;
        int   i0 = 0, i1 = 0;
        float ent = 0.0f;
        float* __restrict__ orow = out_probs + (size_t)gRow * E_DIM;
        for (int e = 0; e < E_DIM; ++e) {
            float p = __expf(L[e] - mx) * inv;
            orow[e] = p;
            ent -= p * __logf(p + EPS_F);
            atomicAdd(&sh_usage[e], p);
            if (p > v0)      { v1 = v0; i1 = i0; v0 = p; i0 = e; }
            else if (p > v1) { v1 = p;  i1 = e; }
        }
        atomicAdd(&sh_ent, ent);

        const float rn = 1.0f / (v0 + v1);
        const int n0 = gRow * 2;
        out_idx_f[n0]     = (float)i0;
        out_idx_f[n0 + 1] = (float)i1;
        idx_i32[n0]     = i0;
        idx_i32[n0 + 1] = i1;
        p2[n0]     = v0 * rn;
        p2[n0 + 1] = v1 * rn;
    }
    __syncthreads();
    if (t < E_DIM) atomicAdd(&usage_sum[t], sh_usage[t]);
    if (t == 0)    atomicAdd(ent_sum, sh_ent);
}

// ---------------- kernel 2: per-block expert histograms (256 slots / block) ----------------
__global__ __launch_bounds__(256)
void rn_cap_count_kernel(const int* __restrict__ idx, int* __restrict__ blkCnt, int nSlots) {
    __shared__ int cnt[E_DIM];
    const int t = threadIdx.x;
    if (t < E_DIM) cnt[t] = 0;
    __syncthreads();
    const int n = blockIdx.x * 256 + t;
    if (n < nSlots) atomicAdd(&cnt[idx[n]], 1);
    __syncthreads();
    if (t < E_DIM) blkCnt[blockIdx.x * E_DIM + t] = cnt[t];
}

// ---------------- kernel 3: per-expert exclusive prefix over blocks ----------------
__global__ void rn_cap_scan_kernel(const int* __restrict__ blkCnt, int* __restrict__ blkPre, int nBlocks) {
    const int e = threadIdx.x;
    if (e >= E_DIM) return;
    int run = 0;
    for (int b = 0; b < nBlocks; ++b) {
        blkPre[b * E_DIM + e] = run;
        run += blkCnt[b * E_DIM + e];
    }
}

// ---------------- kernel 4: FCFS capacity mask + final renorm + confidence ----------------
// 128 threads (tokens) per block; 256 contiguous slots per block.
__global__ __launch_bounds__(128)
void rn_cap_apply_kernel(const int* __restrict__ idx, const int* __restrict__ blkPre,
                         const float* __restrict__ p2, const int* __restrict__ cap_ptr,
                         float* __restrict__ out_topk, float* __restrict__ conf_sum) {
    __shared__ int   sh_e[256];
    __shared__ float sconf;
    const int t = threadIdx.x;
    if (t == 0) sconf = 0.0f;
    const int base = blockIdx.x * 256;
    sh_e[t]       = idx[base + t];
    sh_e[t + 128] = idx[base + t + 128];
    __syncthreads();

    const int cap = cap_ptr[0];
    const int s0 = 2 * t, s1 = 2 * t + 1;
    const int e0 = sh_e[s0], e1 = sh_e[s1];
    int r0 = 0, r1 = 0;
    for (int j = 0; j <= s0; ++j) r0 += (sh_e[j] == e0) ? 1 : 0;
    for (int j = 0; j <= s1; ++j) r1 += (sh_e[j] == e1) ? 1 : 0;
    r0 += blkPre[blockIdx.x * E_DIM + e0];
    r1 += blkPre[blockIdx.x * E_DIM + e1];
    const float m0 = (r0 <= cap) ? 1.0f : 0.0f;
    const float m1 = (r1 <= cap) ? 1.0f : 0.0f;
    float q0 = p2[base + s0] * m0;
    float q1 = p2[base + s1] * m1;
    const float den = q0 + q1 + EPS_F;
    q0 /= den; q1 /= den;
    out_topk[base + s0] = q0;
    out_topk[base + s1] = q1;
    atomicAdd(&sconf, fmaxf(q0, q1));
    __syncthreads();
    if (t == 0) atomicAdd(conf_sum, sconf);
}

// ---------------- kernel 5: scalars ----------------
__global__ void rn_finalize_kernel(const float* __restrict__ usage_sum,
                                   const float* __restrict__ ent_sum,
                                   const float* __restrict__ conf_sum,
                                   float* __restrict__ out_usage,
                                   float* __restrict__ out_lb,
                                   float* __restrict__ out_ent,
                                   float* __restrict__ out_conf,
                                   int nTok) {
    __shared__ float red[E_DIM];
    const int e = threadIdx.x;
    const float invN = 1.0f / (float)nTok;
    if (e < E_DIM) {
        const float u = usage_sum[e] * invN;
        out_usage[e] = u;
        const float tgt = 1.0f / (float)E_DIM;
        red[e] = tgt * (__logf(tgt) - __logf(u));
    }
    __syncthreads();
    if (e == 0) {
        float s = 0.0f;
        for (int i = 0; i < E_DIM; ++i) s += red[i];
        out_lb[0]   = (s / (float)E_DIM) * LBW;
        out_ent[0]  = ent_sum[0] * invN;
        out_conf[0] = conf_sum[0] * invN;
    }
}

// ---------------- launcher ----------------
extern "C" void kernel_launch(void* const* d_in, const int* in_sizes, int n_in,
                              void* d_out, int out_size, void* d_ws, size_t ws_size,
                              hipStream_t stream) {
    const float* H    = (const float*)d_in[0];   // [B,S,D]
    const float* Wm   = (const float*)d_in[1];   // [D,E]
    const float* bias = (const float*)d_in[2];   // [E]
    const int*   cap  = (const int*)d_in[3];     // scalar

    float* out = (float*)d_out;

    const int nTok   = in_sizes[0] / D_DIM;      // 32768
    const int nSlots = nTok * 2;                 // 65536
    const int nCapBlocks = (nSlots + 255) / 256; // 256

    // d_out layout (floats, reference tuple order)
    const size_t OFF_IDX   = 0;
    const size_t OFF_TOPK  = OFF_IDX + (size_t)nSlots;
    const size_t OFF_PROBS = OFF_TOPK + (size_t)nSlots;
    const size_t OFF_LB    = OFF_PROBS + (size_t)nTok * E_DIM;
    const size_t OFF_ENT   = OFF_LB + 1;
    const size_t OFF_USE   = OFF_ENT + 1;
    const size_t OFF_CONF  = OFF_USE + E_DIM;

    // ws carve-up
    char* w = (char*)d_ws;
    float* p2       = (float*)w; w += (size_t)nSlots * sizeof(float);
    int*   idx_i32  = (int*)w;   w += (size_t)nSlots * sizeof(int);
    int*   blkCnt   = (int*)w;   w += (size_t)nCapBlocks * E_DIM * sizeof(int);
    int*   blkPre   = (int*)w;   w += (size_t)nCapBlocks * E_DIM * sizeof(int);
    float* usage_sum = (float*)w; w += E_DIM * sizeof(float);
    float* ent_sum   = (float*)w; w += sizeof(float);
    float* conf_sum  = (float*)w; w += sizeof(float);
    float* WT        = (float*)w; w += (size_t)D_DIM * E_DIM * sizeof(float);

    rn_transpose_w_kernel<<<(D_DIM * E_DIM) / 256, 256, 0, stream>>>(Wm, WT);
    rn_init_kernel<<<1, 128, 0, stream>>>(usage_sum, ent_sum, conf_sum);

    // 128 rows per block -> nTok/128 blocks
    rn_gemm_softmax_kernel<<<nTok / 128, 128, 0, stream>>>(
        H, WT, bias,
        out + OFF_PROBS, out + OFF_IDX,
        idx_i32, p2, usage_sum, ent_sum);

    rn_cap_count_kernel<<<nCapBlocks, 256, 0, stream>>>(idx_i32, blkCnt, nSlots);
    rn_cap_scan_kernel<<<1, 64, 0, stream>>>(blkCnt, blkPre, nCapBlocks);
    rn_cap_apply_kernel<<<nCapBlocks, 128, 0, stream>>>(
        idx_i32, blkPre, p2, cap, out + OFF_TOPK, conf_sum);

    rn_finalize_kernel<<<1, 64, 0, stream>>>(
        usage_sum, ent_sum, conf_sum,
        out + OFF_USE, out + OFF_LB, out + OFF_ENT, out + OFF_CONF, nTok);
}